// SequenceAttentionPoolingLayer_88115549045540
// MI455X (gfx1250) — compile-verified
//
#include <hip/hip_runtime.h>
#include <hip/hip_bf16.h>
#include <math.h>

typedef __attribute__((ext_vector_type(2))) float v2f;
typedef __attribute__((ext_vector_type(8))) float v8f;

#define QDIM 128   // query feature dim
#define HDIM 128   // history feature dim
#define LROWS 200  // history length

// ---------------------------------------------------------------------------
// Kernel 1: proj[b,h] = sum_q query[b,q] * W[q,h]
// M = B (8192), N = HDIM (128), K = QDIM (128), all fp32.
// One wave (32 threads) computes one 16x16 tile of proj via 32 chained
// v_wmma_f32_16x16x4_f32 ops (verified lowering in round 1).
// ---------------------------------------------------------------------------
__global__ void proj_gemm_wmma(const float* __restrict__ Q,
                               const float* __restrict__ W,
                               float* __restrict__ P) {
  const int lane  = threadIdx.x;          // 0..31, one wave per block
  const int tileM = blockIdx.x;           // B/16 tiles
  const int tileN = blockIdx.y;           // HDIM/16 = 8 tiles

  const int m  = tileM * 16 + (lane & 15);
  const int n  = tileN * 16 + (lane & 15);
  const int kh = (lane >> 4) * 2;         // 0 for lanes 0-15, 2 for lanes 16-31

  v8f acc = {};
#pragma unroll
  for (int kk = 0; kk < QDIM; kk += 4) {
    const int ka = kk + kh;               // even -> 8B-aligned float2 load
    v2f a = *(const v2f*)(Q + (size_t)m * QDIM + ka);
    v2f b;
    b.x = W[(size_t)ka * HDIM + n];
    b.y = W[(size_t)(ka + 1) * HDIM + n];
    acc = __builtin_amdgcn_wmma_f32_16x16x4_f32(
        /*neg_a=*/false, a, /*neg_b=*/false, b,
        /*c_mod=*/(short)0, acc, /*reuse_a=*/false, /*reuse_b=*/false);
  }

  const int row0 = tileM * 16 + (lane >> 4) * 8;
  const int col  = tileN * 16 + (lane & 15);
#pragma unroll
  for (int r = 0; r < 8; ++r)
    P[(size_t)(row0 + r) * HDIM + col] = acc[r];
}

// ---------------------------------------------------------------------------
// Kernel 2: per batch row b, stage hist[b] (200x128 f32 = 100 KB) in LDS once
// (hist = 839 MB is the only HBM stream that matters; single pass => ~36 us
// roofline at 23.3 TB/s). Staging uses CDNA5 GLOBAL_LOAD_ASYNC_TO_LDS_B128
// (memory -> LDS direct, ASYNCcnt, no VGPR round-trip). Then: logits =
// hist . proj, faithful mask (l < len -> 1e-9), softmax over L,
// out = score^T . hist from the same LDS tile. 256 threads = 8 wave32.
// ---------------------------------------------------------------------------
#if defined(__has_builtin)
#if __has_builtin(__builtin_amdgcn_global_load_async_to_lds_b128)
#define USE_ASYNC_LDS 1
#endif
#endif

// Match the builtin's parameter types exactly (from the round-2 diagnostic:
// pointee is 'int __attribute__((vector_size(4*sizeof(int))))').
typedef int v4i_mem __attribute__((vector_size(4 * sizeof(int))));
typedef __attribute__((address_space(1))) v4i_mem g_v4i;  // global src
typedef __attribute__((address_space(3))) v4i_mem l_v4i;  // LDS dst

__global__ void attn_pool(const float* __restrict__ hist,
                          const int*   __restrict__ len,
                          const float* __restrict__ proj,
                          float* __restrict__ out) {
  extern __shared__ float smem[];
  float* hist_s  = smem;                         // LROWS*HDIM = 25600 floats
  float* proj_s  = hist_s + LROWS * HDIM;        // 128
  float* score_s = proj_s + HDIM;                // 200
  float* red     = score_s + LROWS;              // 256

  const int b = blockIdx.x;
  const int t = threadIdx.x;                     // 0..255
  const int wave = t >> 5;
  const int lane = t & 31;

  // --- stage hist tile into LDS ---
  const float4* hb4 = (const float4*)(hist + (size_t)b * LROWS * HDIM);
  float4* hs4 = (float4*)hist_s;
#ifdef USE_ASYNC_LDS
#pragma unroll 5
  for (int i = t; i < LROWS * HDIM / 4; i += 256) {
    __builtin_amdgcn_global_load_async_to_lds_b128(
        (g_v4i*)(hb4 + i), (l_v4i*)(hs4 + i),
        /*offset=*/0, /*cpol=*/0);
  }
#if __has_builtin(__builtin_amdgcn_s_wait_asynccnt)
  __builtin_amdgcn_s_wait_asynccnt(0);
#else
  asm volatile("s_wait_asynccnt 0x0" ::: "memory");
#endif
#else
#pragma unroll 5
  for (int i = t; i < LROWS * HDIM / 4; i += 256) hs4[i] = hb4[i];
#endif
  if (t < HDIM) proj_s[t] = proj[(size_t)b * HDIM + t];
  __syncthreads();

  // --- logits: each wave owns rows l = wave, wave+8, ... (25 rows/wave) ---
  const int Lb = len[b];
  for (int l = wave; l < LROWS; l += 8) {
    const float* hr = hist_s + l * HDIM;
    float p = 0.f;
#pragma unroll
    for (int h = lane; h < HDIM; h += 32) p = fmaf(hr[h], proj_s[h], p);
#pragma unroll
    for (int off = 16; off > 0; off >>= 1) p += __shfl_down(p, off, 32);
    if (lane == 0) score_s[l] = (l < Lb) ? 1e-9f : p;  // faithful to reference
  }
  __syncthreads();

  // --- softmax over 200 logits (wave 0 only; ~7 elems/lane) ---
  if (wave == 0) {
    float m = -INFINITY;
    for (int l = lane; l < LROWS; l += 32) m = fmaxf(m, score_s[l]);
#pragma unroll
    for (int off = 16; off > 0; off >>= 1) m = fmaxf(m, __shfl_xor(m, off, 32));
    float s = 0.f;
    for (int l = lane; l < LROWS; l += 32) {
      float e = __expf(score_s[l] - m);
      score_s[l] = e;
      s += e;
    }
#pragma unroll
    for (int off = 16; off > 0; off >>= 1) s += __shfl_xor(s, off, 32);
    const float inv = __frcp_rn(s);
    for (int l = lane; l < LROWS; l += 32) score_s[l] *= inv;
  }
  __syncthreads();

  // --- weighted sum from LDS (second use of the staged tile, zero HBM) ---
  const int h = t & (HDIM - 1);
  const int half = t >> 7;                       // 0 or 1 -> rows [0,100)/[100,200)
  float acc = 0.f;
#pragma unroll 4
  for (int l = half * 100; l < half * 100 + 100; ++l)
    acc = fmaf(score_s[l], hist_s[l * HDIM + h], acc);
  red[t] = acc;
  __syncthreads();
  if (t < HDIM) out[(size_t)b * HDIM + t] = red[t] + red[t + HDIM];
}

// ---------------------------------------------------------------------------
extern "C" void kernel_launch(void* const* d_in, const int* in_sizes, int n_in,
                              void* d_out, int out_size, void* d_ws, size_t ws_size,
                              hipStream_t stream) {
  const float* query = (const float*)d_in[0];      // [B, 128] f32
  const float* hist  = (const float*)d_in[1];      // [B, 200, 128] f32
  const int*   lens  = (const int*)d_in[2];        // [B] int32 (JAX x64 off)
  const float* W     = (const float*)d_in[3];      // [128, 128] f32
  float* outp = (float*)d_out;                     // [B, 128] f32

  const int B = in_sizes[2];                       // 8192
  float* proj = (float*)d_ws;                      // [B, 128] f32 scratch

  dim3 g1(B / 16, HDIM / 16);
  proj_gemm_wmma<<<g1, 32, 0, stream>>>(query, W, proj);

  const size_t shbytes = (size_t)(LROWS * HDIM + HDIM + LROWS + 256) * sizeof(float);
  attn_pool<<<B, 256, shbytes, stream>>>(hist, lens, proj, outp);
}